// GIN_85452669321990
// MI455X (gfx1250) — compile-verified
//
#include <hip/hip_runtime.h>
#include <hip/hip_bf16.h>

typedef float v2f __attribute__((ext_vector_type(2)));
typedef float v8f __attribute__((ext_vector_type(8)));

#define N_NODES   100000
#define N_EDGES   3200000
#define HIDDEN    128
#define N_LAYERS  3
#define N_CLASSES 10
#define N_GRAPHS  512
#define BN_EPS    1e-5f

#define CH      512
#define NCHUNK  ((N_NODES + CH - 1) / CH)   // 196

// ---------------- CSR build ----------------

__global__ void k_hist(const int* __restrict__ dst, int* __restrict__ counts) {
    int e = blockIdx.x * blockDim.x + threadIdx.x;
    if (e < N_EDGES) atomicAdd(&counts[dst[e]], 1);
}

__global__ void k_chunksum(const int* __restrict__ counts, int* __restrict__ partials) {
    int b = blockIdx.x * blockDim.x + threadIdx.x;
    if (b < NCHUNK) {
        int base = b * CH;
        int end  = base + CH; if (end > N_NODES) end = N_NODES;
        int s = 0;
        for (int i = base; i < end; ++i) s += counts[i];
        partials[b] = s;
    }
}

__global__ void k_scanpart(int* __restrict__ partials) {
    if (blockIdx.x == 0 && threadIdx.x == 0) {
        int run = 0;
        for (int i = 0; i < NCHUNK; ++i) { int v = partials[i]; partials[i] = run; run += v; }
    }
}

__global__ void k_fillptr(const int* __restrict__ counts, const int* __restrict__ partials,
                          int* __restrict__ row_ptr, int* __restrict__ nxt) {
    int b = blockIdx.x * blockDim.x + threadIdx.x;
    if (b < NCHUNK) {
        int run  = partials[b];
        int base = b * CH;
        int end  = base + CH; if (end > N_NODES) end = N_NODES;
        for (int i = base; i < end; ++i) {
            row_ptr[i] = run; nxt[i] = run; run += counts[i];
        }
    }
    if (b == 0) row_ptr[N_NODES] = N_EDGES;
}

__global__ void k_fill(const int* __restrict__ src, const int* __restrict__ dst,
                       int* __restrict__ nxt, int* __restrict__ colIdx) {
    int e = blockIdx.x * blockDim.x + threadIdx.x;
    if (e < N_EDGES) {
        int pos = atomicAdd(&nxt[dst[e]], 1);
        colIdx[pos] = src[e];
    }
}

// ---------------- aggregation: z = (1+eps)*h + sum_{nbr} h[nbr] ----------------
// one wave (32 lanes) per node; each lane owns 4 contiguous channels (float4).
// each neighbor load is a coalesced 512B wave-wide read that hits L2 (h = 51MB << 192MB L2).

__global__ void __launch_bounds__(256) k_agg(const float* __restrict__ h,
                                             const int* __restrict__ row_ptr,
                                             const int* __restrict__ colIdx,
                                             const float* __restrict__ eps_p,
                                             float* __restrict__ z) {
    int wave = (blockIdx.x * blockDim.x + threadIdx.x) >> 5;
    int lane = threadIdx.x & 31;
    if (wave >= N_NODES) return;
    float e1 = 1.0f + *eps_p;
    const float4* hv = (const float4*)h;
    float4 hc = hv[(size_t)wave * 32 + lane];
    float4 acc = make_float4(e1 * hc.x, e1 * hc.y, e1 * hc.z, e1 * hc.w);
    int beg = row_ptr[wave], end = row_ptr[wave + 1];
    for (int j = beg; j < end; ++j) {
        int s = colIdx[j];
        float4 v = hv[(size_t)s * 32 + lane];
        acc.x += v.x; acc.y += v.y; acc.z += v.z; acc.w += v.w;
    }
    ((float4*)z)[(size_t)wave * 32 + lane] = acc;
}

// ---------------- fused MLP + BN per layer, fp32 WMMA 16x16x4 ----------------
// block = 256 threads = 8 waves; M-tile = 64 rows, full N = 128.
// wave w: row-tile rt = w/2 (16 rows), col-tiles (w%2)*4 .. +3 (64 cols).

#define TM    64
#define LDP   132   // LDS row stride (floats); 132%64=4 -> conflict-free strided row access

__global__ void __launch_bounds__(256) k_mlp(
    const float* __restrict__ Zin,
    const float* __restrict__ W1, const float* __restrict__ B1,
    const float* __restrict__ W2, const float* __restrict__ B2,
    const float* __restrict__ gamma, const float* __restrict__ beta,
    const float* __restrict__ mu,    const float* __restrict__ var,
    float* __restrict__ Hout) {
    __shared__ float As[TM * LDP];
    __shared__ float Zs[TM * LDP];
    const int m0 = blockIdx.x * TM;
    const int t  = threadIdx.x;

    // load 64x128 A tile (zero-fill OOB rows); float4 per thread x 8
#pragma unroll
    for (int i = 0; i < 8; ++i) {
        int f   = t + i * 256;        // 0..2047 float4 slots
        int row = f >> 5;
        int c4  = f & 31;
        float4 v = make_float4(0.f, 0.f, 0.f, 0.f);
        int gr = m0 + row;
        if (gr < N_NODES) v = ((const float4*)Zin)[(size_t)gr * 32 + c4];
        *(float4*)&As[row * LDP + c4 * 4] = v;
    }
    __syncthreads();

    const int w      = t >> 5;
    const int lane   = t & 31;
    const int rt     = w >> 1;          // 0..3
    const int ntBase = (w & 1) * 4;     // 0 or 4
    const int halfK  = lane >> 4;       // 0,1
    const int mrow   = lane & 15;       // A row / B,D col within tile

    // ---- GEMM1: Z1 = relu(A @ W1 + b1) ----
    v8f acc[4] = {};
    for (int kc = 0; kc < 32; ++kc) {
        int k = kc * 4 + 2 * halfK;
        v2f a = *(const v2f*)&As[(rt * 16 + mrow) * LDP + k];
#pragma unroll
        for (int i = 0; i < 4; ++i) {
            int cb = (ntBase + i) * 16 + mrow;
            v2f b;
            b.x = W1[k * HIDDEN + cb];
            b.y = W1[(k + 1) * HIDDEN + cb];
            acc[i] = __builtin_amdgcn_wmma_f32_16x16x4_f32(
                false, a, false, b, (short)0, acc[i], false, false);
        }
    }
#pragma unroll
    for (int i = 0; i < 4; ++i) {
        int cb = (ntBase + i) * 16 + mrow;
        float bias = B1[cb];
#pragma unroll
        for (int v = 0; v < 8; ++v) {
            int r = rt * 16 + v + 8 * halfK;
            float x = acc[i][v] + bias;
            Zs[r * LDP + cb] = x > 0.f ? x : 0.f;
        }
    }
    __syncthreads();

    // ---- GEMM2: H = BN(relu(Z1 @ W2 + b2)) ----
    v8f acc2[4] = {};
    for (int kc = 0; kc < 32; ++kc) {
        int k = kc * 4 + 2 * halfK;
        v2f a = *(const v2f*)&Zs[(rt * 16 + mrow) * LDP + k];
#pragma unroll
        for (int i = 0; i < 4; ++i) {
            int cb = (ntBase + i) * 16 + mrow;
            v2f b;
            b.x = W2[k * HIDDEN + cb];
            b.y = W2[(k + 1) * HIDDEN + cb];
            acc2[i] = __builtin_amdgcn_wmma_f32_16x16x4_f32(
                false, a, false, b, (short)0, acc2[i], false, false);
        }
    }
#pragma unroll
    for (int i = 0; i < 4; ++i) {
        int cb   = (ntBase + i) * 16 + mrow;
        float bias = B2[cb];
        float sc = gamma[cb] * rsqrtf(var[cb] + BN_EPS);
        float sh = beta[cb] - mu[cb] * sc;
#pragma unroll
        for (int v = 0; v < 8; ++v) {
            int r  = rt * 16 + v + 8 * halfK;
            int gr = m0 + r;
            float x = acc2[i][v] + bias;
            x = x > 0.f ? x : 0.f;
            x = x * sc + sh;
            if (gr < N_NODES) Hout[(size_t)gr * HIDDEN + cb] = x;
        }
    }
}

// ---------------- pooling ----------------

__global__ void __launch_bounds__(256) k_poolsum(const float* __restrict__ h,
                                                 const int* __restrict__ batch,
                                                 float* __restrict__ pooled) {
    int tid  = blockIdx.x * blockDim.x + threadIdx.x;
    int node = tid >> 5;
    int lane = tid & 31;
    if (node >= N_NODES) return;
    int g = batch[node];
    float4 v = ((const float4*)h)[(size_t)node * 32 + lane];
    float* p = &pooled[g * HIDDEN + lane * 4];
    atomicAdd(p + 0, v.x); atomicAdd(p + 1, v.y);
    atomicAdd(p + 2, v.z); atomicAdd(p + 3, v.w);
}

__global__ void k_count(const int* __restrict__ batch, float* __restrict__ cnt) {
    int i = blockIdx.x * blockDim.x + threadIdx.x;
    if (i < N_NODES) atomicAdd(&cnt[batch[i]], 1.0f);
}

// ---------------- head: relu(pooled@lin1+b) @ lin2 + b -> log_softmax ----------------

__global__ void __launch_bounds__(128) k_head(const float* __restrict__ pooled,
                                              const float* __restrict__ cnt,
                                              const float* __restrict__ l1w,
                                              const float* __restrict__ l1b,
                                              const float* __restrict__ l2w,
                                              const float* __restrict__ l2b,
                                              float* __restrict__ out) {
    __shared__ float p[HIDDEN];
    __shared__ float t1[HIDDEN];
    __shared__ float lg[16];
    int g = blockIdx.x, c = threadIdx.x;
    float cn  = cnt[g];
    float inv = 1.0f / (cn > 1.0f ? cn : 1.0f);
    p[c] = pooled[g * HIDDEN + c] * inv;
    __syncthreads();
    float s = l1b[c];
    for (int k = 0; k < HIDDEN; ++k) s += p[k] * l1w[k * HIDDEN + c];
    t1[c] = s > 0.f ? s : 0.f;
    __syncthreads();
    if (c < N_CLASSES) {
        float s2 = l2b[c];
        for (int k = 0; k < HIDDEN; ++k) s2 += t1[k] * l2w[k * N_CLASSES + c];
        lg[c] = s2;
    }
    __syncthreads();
    if (c < N_CLASSES) {
        float m = lg[0];
        for (int j = 1; j < N_CLASSES; ++j) m = lg[j] > m ? lg[j] : m;
        float se = 0.f;
        for (int j = 0; j < N_CLASSES; ++j) se += expf(lg[j] - m);
        out[g * N_CLASSES + c] = lg[c] - m - logf(se);
    }
}

// ---------------- host ----------------

extern "C" void kernel_launch(void* const* d_in, const int* in_sizes, int n_in,
                              void* d_out, int out_size, void* d_ws, size_t ws_size,
                              hipStream_t stream) {
    const float* x     = (const float*)d_in[0];
    const int*   ei    = (const int*)d_in[1];
    const int*   batch = (const int*)d_in[2];
    const float* W1    = (const float*)d_in[3];
    const float* b1    = (const float*)d_in[4];
    const float* W2    = (const float*)d_in[5];
    const float* b2    = (const float*)d_in[6];
    const float* gamma = (const float*)d_in[7];
    const float* beta  = (const float*)d_in[8];
    const float* mu    = (const float*)d_in[9];
    const float* var   = (const float*)d_in[10];
    const float* eps   = (const float*)d_in[11];
    const float* l1w   = (const float*)d_in[12];
    const float* l1b   = (const float*)d_in[13];
    const float* l2w   = (const float*)d_in[14];
    const float* l2b   = (const float*)d_in[15];

    const int* src = ei;
    const int* dst = ei + N_EDGES;

    char*  ws  = (char*)d_ws;
    size_t off = 0;
    auto alloc = [&](size_t bytes) {
        void* p = ws + off;
        off = (off + bytes + 255) & ~(size_t)255;
        return p;
    };
    float* H       = (float*)alloc((size_t)N_NODES * HIDDEN * 4);
    float* Z       = (float*)alloc((size_t)N_NODES * HIDDEN * 4);
    int*   row_ptr = (int*)  alloc((size_t)(N_NODES + 1) * 4);
    int*   colIdx  = (int*)  alloc((size_t)N_EDGES * 4);
    int*   counts  = (int*)  alloc((size_t)N_NODES * 4);
    int*   nxt     = (int*)  alloc((size_t)N_NODES * 4);
    int*   parts   = (int*)  alloc(1024 * 4);
    float* pooled  = (float*)alloc((size_t)N_GRAPHS * HIDDEN * 4);
    float* cnt     = (float*)alloc((size_t)N_GRAPHS * 4);

    hipMemsetAsync(counts, 0, (size_t)N_NODES * 4, stream);
    hipMemsetAsync(pooled, 0, (size_t)N_GRAPHS * HIDDEN * 4, stream);
    hipMemsetAsync(cnt,    0, (size_t)N_GRAPHS * 4, stream);

    // CSR build (once; reused for all 3 layers)
    k_hist    <<<(N_EDGES + 255) / 256, 256, 0, stream>>>(dst, counts);
    k_chunksum<<<(NCHUNK + 255) / 256, 256, 0, stream>>>(counts, parts);
    k_scanpart<<<1, 1, 0, stream>>>(parts);
    k_fillptr <<<(NCHUNK + 255) / 256, 256, 0, stream>>>(counts, parts, row_ptr, nxt);
    k_fill    <<<(N_EDGES + 255) / 256, 256, 0, stream>>>(src, dst, nxt, colIdx);

    const int aggBlocks = (N_NODES * 32) / 256;            // 12500
    const int mlpBlocks = (N_NODES + TM - 1) / TM;         // 1563

    for (int l = 0; l < N_LAYERS; ++l) {
        const float* hin = (l == 0) ? x : H;
        k_agg<<<aggBlocks, 256, 0, stream>>>(hin, row_ptr, colIdx, eps + l, Z);
        k_mlp<<<mlpBlocks, 256, 0, stream>>>(
            Z,
            W1 + (size_t)l * HIDDEN * HIDDEN, b1 + (size_t)l * HIDDEN,
            W2 + (size_t)l * HIDDEN * HIDDEN, b2 + (size_t)l * HIDDEN,
            gamma + (size_t)l * HIDDEN, beta + (size_t)l * HIDDEN,
            mu + (size_t)l * HIDDEN,    var + (size_t)l * HIDDEN,
            H);
    }

    k_poolsum<<<aggBlocks, 256, 0, stream>>>(H, batch, pooled);
    k_count  <<<(N_NODES + 255) / 256, 256, 0, stream>>>(batch, cnt);
    k_head   <<<N_GRAPHS, 128, 0, stream>>>(pooled, cnt, l1w, l1b, l2w, l2b, (float*)d_out);
}